// Spiral_12601434046976
// MI455X (gfx1250) — compile-verified
//
#include <hip/hip_runtime.h>
#include <math.h>

#define DEV_INLINE __device__ __forceinline__

namespace {

constexpr int   B_    = 8;
constexpr int   L_    = 1000000;
constexpr int   SIZE  = 1349;               // odd spiral grid side for L=1e6
constexpr int   NC    = SIZE * SIZE;        // 1,819,801 grid cells
constexpr int   NRING = 1024;               // max ring id = round(674*sqrt(2)) = 953
constexpr float PI_F  = 3.14159274101257324f; // float32(np.pi) = 0x40490FDB
constexpr float HALF  = 0.5f * (float)SIZE - 0.5f; // 674.0 -> integer coords

typedef unsigned long long u64;
typedef __attribute__((ext_vector_type(4))) float v4f;
typedef __attribute__((ext_vector_type(2))) int   v2i;

// ---- exact fp32 reproduction of the reference's spiral key ------------------
DEV_INLINE float ring_radius(int g) {
  int i = g / SIZE, j = g - i * SIZE;       // x1 = rnge[j], x2 = rnge[i] ('xy' meshgrid)
  float x1 = (float)j - HALF;
  float x2 = (float)i - HALF;
  float r2 = x1 * x1 + x2 * x2;             // exact: integer < 2^24
  return (float)sqrt((double)r2);           // correctly-rounded fp32 sqrt
}

// key = fl32(round(r)*2*PI + phi); all keys >= 0, so key-bits are order-preserving
// as u32. pair = (keybits << 32) | cellIdx == stable (key, idx) lexicographic order.
DEV_INLINE u64 spiral_pair(int g, int* ringOut) {
  int i = g / SIZE, j = g - i * SIZE;
  float x1 = (float)j - HALF;
  float x2 = (float)i - HALF;
  float r2 = x1 * x1 + x2 * x2;
  float r  = (float)sqrt((double)r2);
  float q  = __fdiv_rn(x1, r);              // NaN only at the center cell (0/0)
  float phi = (float)acos((double)q);       // correctly-rounded fp32 acos
  if (isnan(phi)) phi = 0.0f;
  float sg = (x2 > 0.0f) ? 1.0f : ((x2 < 0.0f) ? -1.0f : 0.0f); // jnp.sign
  phi = __fmul_rn(phi, sg);
  if ((x2 == 0.0f) && (x1 < 0.0f)) phi = __fadd_rn(phi, PI_F);
  float rk  = rintf(r);                     // jnp.round; never lands on .5 boundary
  float key = __fadd_rn(__fmul_rn(__fmul_rn(rk, 2.0f), PI_F), phi);
  *ringOut = (int)rk;
  unsigned int kb = __float_as_uint(key);   // key >= 0 -> monotone bit pattern
  return ((u64)kb << 32) | (unsigned int)g;
}

// ---- CDNA5 async global->LDS staging ---------------------------------------
#if defined(__gfx1250__) && __has_builtin(__builtin_amdgcn_global_load_async_to_lds_b64)
#define HAVE_ASYNC_B64 1
typedef __attribute__((address_space(1))) v2i g_v2i_t;   // global
typedef __attribute__((address_space(3))) v2i l_v2i_t;   // LDS
#endif

DEV_INLINE void async_g2l_b64(const void* gsrc, void* ldst) {
#if defined(HAVE_ASYNC_B64)
  __builtin_amdgcn_global_load_async_to_lds_b64(
      (g_v2i_t*)(gsrc), (l_v2i_t*)(ldst), 0, 0);
#else
  *(u64*)ldst = *(const u64*)gsrc;          // fallback through VGPRs
#endif
}

DEV_INLINE void wait_async0() {
#if defined(__gfx1250__) && __has_builtin(__builtin_amdgcn_s_wait_asynccnt)
  __builtin_amdgcn_s_wait_asynccnt(0);
#elif defined(__gfx1250__)
  asm volatile("s_wait_asynccnt 0" ::: "memory");
#endif
}

// ---- kernels ---------------------------------------------------------------
__global__ void k_zero_rings(int* __restrict__ ringCount) {
  if (threadIdx.x < NRING) ringCount[threadIdx.x] = 0;
}

__global__ void k_count(int* __restrict__ ringCount) {
  int g = blockIdx.x * blockDim.x + threadIdx.x;
  if (g >= NC) return;
  int k = (int)rintf(ring_radius(g));
  atomicAdd(&ringCount[k], 1);
}

__global__ void k_scan(const int* __restrict__ ringCount,
                       int* __restrict__ ringOffset,
                       int* __restrict__ ringCursor) {
  if (threadIdx.x == 0 && blockIdx.x == 0) {
    int s = 0;
    for (int i = 0; i < NRING; ++i) {
      ringOffset[i] = s; ringCursor[i] = s; s += ringCount[i];
    }
  }
}

__global__ void k_bin(int* __restrict__ ringCursor, u64* __restrict__ pairArr) {
  int g = blockIdx.x * blockDim.x + threadIdx.x;
  if (g >= NC) return;
  int k;
  u64 pair = spiral_pair(g, &k);
  int p = atomicAdd(&ringCursor[k], 1);
  pairArr[p] = pair;
}

// One block per ring: async-DMA the ring's (key,idx) u64 pairs into LDS,
// bitonic-sort them (unique keys -> stability irrelevant), emit ranks.
constexpr int ST = 1024;     // sort threads (32 wave32)
__global__ void __launch_bounds__(ST)
k_rank(const int* __restrict__ ringCount, const int* __restrict__ ringOffset,
       const u64* __restrict__ pairArr, int* __restrict__ rankArr) {
  extern __shared__ u64 s[];                // 8192 * 8B = 64 KB dynamic LDS
  const int ring  = blockIdx.x;
  const int n     = ringCount[ring];        // max ~4300 (<= 8192)
  const int start = ringOffset[ring];
  if (n == 0) return;
  int m = 1; while (m < n) m <<= 1;         // power-of-two padded length
  for (int t = (int)threadIdx.x; t < n; t += ST)
    async_g2l_b64(&pairArr[start + t], &s[t]);
  for (int t = n + (int)threadIdx.x; t < m; t += ST)
    s[t] = ~0ull;                           // +inf sentinel (all real pairs smaller)
  wait_async0();                            // drain this wave's ASYNCcnt
  __syncthreads();
  for (int k = 2; k <= m; k <<= 1) {
    for (int j = k >> 1; j > 0; j >>= 1) {
      for (int i = (int)threadIdx.x; i < m; i += ST) {
        const int l = i ^ j;
        if (l > i) {
          const u64 a = s[i], b = s[l];
          const bool up = ((i & k) == 0);
          if (up ? (a > b) : (a < b)) { s[i] = b; s[l] = a; }
        }
      }
      __syncthreads();
    }
  }
  for (int p = (int)threadIdx.x; p < n; p += ST)
    rankArr[(int)(s[p] & 0xFFFFFFFFull)] = start + p;
}

// Hot kernel (~361 MB @ 23.3 TB/s ~= 15.5 us): coalesced NT float4 stores,
// rank-indexed 16B gathers served out of L2 (input fits in 192 MB L2).
__global__ void __launch_bounds__(256)
k_gather(const v4f* __restrict__ in, const int* __restrict__ rankArr,
         v4f* __restrict__ out) {
  const int g = blockIdx.x * 256 + (int)threadIdx.x;
  if (g >= NC) return;
  const int  r   = rankArr[g];
  const bool hit = (r < L_);
#pragma unroll
  for (int b = 0; b < B_; ++b) {
    v4f v = {};
    if (hit) v = in[(long)b * L_ + r];                  // RT load: keep in L2
    __builtin_nontemporal_store(v, &out[(long)b * NC + g]); // NT: write-once stream
  }
}

} // anonymous namespace

extern "C" void kernel_launch(void* const* d_in, const int* in_sizes, int n_in,
                              void* d_out, int out_size, void* d_ws, size_t ws_size,
                              hipStream_t stream) {
  (void)in_sizes; (void)n_in; (void)out_size; (void)ws_size;

  const v4f* in  = (const v4f*)d_in[0];
  v4f*       out = (v4f*)d_out;

  // Workspace layout (~22 MB total)
  char* ws = (char*)d_ws;
  int* ringCount  = (int*)(ws);
  int* ringOffset = (int*)(ws + 4096);
  int* ringCursor = (int*)(ws + 8192);
  size_t off = 12288;
  u64* pairArr = (u64*)(ws + off);
  off += (((size_t)NC * 8) + 255) & ~(size_t)255;
  int* rankArr = (int*)(ws + off);

  const int gridNC = (NC + 255) / 256;

  k_zero_rings<<<1, NRING, 0, stream>>>(ringCount);
  k_count<<<gridNC, 256, 0, stream>>>(ringCount);
  k_scan<<<1, 1, 0, stream>>>(ringCount, ringOffset, ringCursor);
  k_bin<<<gridNC, 256, 0, stream>>>(ringCursor, pairArr);
  k_rank<<<NRING, ST, 8192 * sizeof(u64), stream>>>(ringCount, ringOffset, pairArr, rankArr);
  k_gather<<<gridNC, 256, 0, stream>>>(in, rankArr, out);
}